// LJModel_70171175682200
// MI455X (gfx1250) — compile-verified
//
#include <hip/hip_runtime.h>

// Module constants from the reference
#define N_MOL_C    1000
#define NB         512    // stage-1 blocks == partials per molecule
#define CUTOFF_F   2.5f
#define CUT_LO_F   2.0f   // CUTOFF - HEAL
#define INV_HEAL_F 2.0f   // 1 / HEAL
#define HALF_DEPTH 0.5f   // WELL_DEPTH * 0.5

typedef float v2f __attribute__((ext_vector_type(2)));
typedef float v8f __attribute__((ext_vector_type(8)));

__global__ void lj_zero_kernel(float* __restrict__ out, int n) {
    int i = blockIdx.x * blockDim.x + threadIdx.x;
    if (i < n) out[i] = 0.0f;
}

__device__ __forceinline__ float lj_edge_energy(float r2) {
    // One v_rsq_f32 replaces sqrt + IEEE divide (~1-2 ulp, fine at fp32 tol)
    float rsq = __builtin_amdgcn_rsqf(r2);
    float r   = r2 * rsq;          // sqrt(r2)
    float ir2 = rsq * rsq;         // 1/r2
    // CustomCutoff switching function
    float t  = (r - CUT_LO_F) * INV_HEAL_F;
    float rf = 1.0f + t * t * (2.0f * t - 3.0f);
    float sw = (r > CUT_LO_F) ? rf : 1.0f;
    sw       = (r > CUTOFF_F) ? 0.0f : sw;
    float p6  = ir2 * ir2 * ir2;   // (R_EQ/r)^6, R_EQ = 1
    float p12 = p6 * p6;
    return (p12 - p6) * sw;
}

__global__ __launch_bounds__(512) void lj_seg_kernel(
    const float4* __restrict__ vec4,    // vec_ij as float4 (3 per 4 edges)
    const float*  __restrict__ vec,     // scalar view for tail
    const int4*   __restrict__ idx4,    // idx_i as int4
    const int*    __restrict__ idx_i,   // scalar view for tail
    const int*    __restrict__ idx_m,   // atom -> molecule (L2-resident, 4 MB)
    float*        __restrict__ partial, // [padded_rows][NB] transposed partials (wmma mode)
    float*        __restrict__ out,     // [N_MOL] accumulators (atomic fallback)
    int n_chunks,                       // n_edges / 4
    int n_edges,
    int wmma_mode)                      // 1: store partials; 0: atomic flush
{
    __shared__ float bins[N_MOL_C];
    for (int t = threadIdx.x; t < N_MOL_C; t += blockDim.x) bins[t] = 0.0f;
    __syncthreads();

    const int gid   = blockIdx.x * blockDim.x + threadIdx.x;
    const int gsize = gridDim.x * blockDim.x;

    // Main vectorized loop: 4 edges / thread / iteration
    for (int c = gid; c < n_chunks; c += gsize) {
        size_t b = 3u * (size_t)c;
        // Prefetch one grid-stride ahead (speculative; OOB silently dropped)
        __builtin_prefetch(&vec4[b + 3u * (size_t)gsize], 0, 0);
        __builtin_prefetch(&idx4[(size_t)c + (size_t)gsize], 0, 0);

        float4 v0 = vec4[b + 0];
        float4 v1 = vec4[b + 1];
        float4 v2 = vec4[b + 2];
        int4   ii = idx4[c];

        float r2[4];
        r2[0] = v0.x * v0.x + v0.y * v0.y + v0.z * v0.z;
        r2[1] = v0.w * v0.w + v1.x * v1.x + v1.y * v1.y;
        r2[2] = v1.z * v1.z + v1.w * v1.w + v2.x * v2.x;
        r2[3] = v2.y * v2.y + v2.z * v2.z + v2.w * v2.w;
        int atom[4] = { ii.x, ii.y, ii.z, ii.w };

#pragma unroll
        for (int k = 0; k < 4; ++k) {
            float yij = lj_edge_energy(r2[k]);
            int   m   = idx_m[atom[k]];           // gather, L2 hit
            atomicAdd(&bins[m], yij);             // ds_add_f32
        }
    }

    // Scalar tail (n_edges % 4)
    for (int e = 4 * n_chunks + gid; e < n_edges; e += gsize) {
        size_t b = 3u * (size_t)e;
        float x = vec[b + 0], y = vec[b + 1], z = vec[b + 2];
        float yij = lj_edge_energy(x * x + y * y + z * z);
        int   m   = idx_m[idx_i[e]];
        atomicAdd(&bins[m], yij);
    }

    __syncthreads();

    if (wmma_mode) {
        // Deterministic path: store block histogram transposed, P[m][block]
        for (int t = threadIdx.x; t < N_MOL_C; t += blockDim.x)
            partial[(size_t)t * NB + blockIdx.x] = bins[t];
    } else {
        // Fallback: atomic flush (tiny vs. the 320 MB stream)
        for (int t = threadIdx.x; t < N_MOL_C; t += blockDim.x) {
            float v = bins[t];
            if (v != 0.0f) atomicAdd(&out[t], HALF_DEPTH * v);
        }
    }
}

// Sum NB partials per molecule with v_wmma_f32_16x16x4_f32.
// One wave handles 16 molecules (rows). A[16x4] holds P[row][k0..k0+3] per the
// ISA f32 A-layout (lanes 0-15: K=k0,k0+1; lanes 16-31: K=k0+2,k0+3).
// B = all-ones, so D = A*1 + C accumulates row sums in every column.
__global__ __launch_bounds__(32) void lj_reduce_wmma(
    const float* __restrict__ partial, float* __restrict__ out, int n_mol)
{
    const int lane = threadIdx.x;            // 0..31, one wave
    const int m0   = blockIdx.x * 16;
    const int row  = m0 + (lane & 15);
    const int koff = (lane >> 4) << 1;       // 0 for lanes 0-15, 2 for 16-31
    const float* base = partial + (size_t)row * NB + koff;

    v8f acc  = {};
    v2f ones = {1.0f, 1.0f};
#pragma unroll 4
    for (int k0 = 0; k0 < NB; k0 += 4) {
        v2f a = *(const v2f*)(base + k0);    // global_load_b64, 8B-aligned
        acc = __builtin_amdgcn_wmma_f32_16x16x4_f32(
            /*neg_a=*/false, a, /*neg_b=*/false, ones,
            /*c_mod=*/(short)0, acc, /*reuse_a=*/false, /*reuse_b=*/false);
    }

    // D layout: lane 0 VGPR v = D[v][0] = rowsum(v); lane 16 VGPR v = D[8+v][0]
    if (lane == 0 || lane == 16) {
        int mbase = m0 + ((lane >> 4) << 3);
#pragma unroll
        for (int v = 0; v < 8; ++v) {
            int m = mbase + v;
            if (m < n_mol) out[m] = HALF_DEPTH * acc[v];
        }
    }
}

extern "C" void kernel_launch(void* const* d_in, const int* in_sizes, int n_in,
                              void* d_out, int out_size, void* d_ws, size_t ws_size,
                              hipStream_t stream) {
    // setup_inputs order: vec_ij [E,3] f32, positions [N,3] f32 (unused),
    //                     idx_i [E] i32, idx_m [N] i32
    const float* vec   = (const float*)d_in[0];
    const int*   idx_i = (const int*)d_in[2];
    const int*   idx_m = (const int*)d_in[3];
    float*       out   = (float*)d_out;

    const int n_edges  = in_sizes[2];
    const int n_chunks = n_edges / 4;
    const int threads  = 512;               // 16 waves of 32 on CDNA5

    const int    red_blocks  = (out_size + 15) / 16;
    const size_t padded_rows = (size_t)red_blocks * 16;
    const size_t ws_need     = padded_rows * NB * sizeof(float);
    const bool   use_wmma    = (out_size <= N_MOL_C) && (ws_size >= ws_need);

    if (use_wmma) {
        lj_seg_kernel<<<NB, threads, 0, stream>>>(
            (const float4*)vec, vec, (const int4*)idx_i, idx_i, idx_m,
            (float*)d_ws, out, n_chunks, n_edges, /*wmma_mode=*/1);
        lj_reduce_wmma<<<red_blocks, 32, 0, stream>>>(
            (const float*)d_ws, out, out_size);
    } else {
        lj_zero_kernel<<<(out_size + 255) / 256, 256, 0, stream>>>(out, out_size);
        lj_seg_kernel<<<512, threads, 0, stream>>>(
            (const float4*)vec, vec, (const int4*)idx_i, idx_i, idx_m,
            nullptr, out, n_chunks, n_edges, /*wmma_mode=*/0);
    }
}